// GenuineLeechLattice_12592844112100
// MI455X (gfx1250) — compile-verified
//
#include <hip/hip_runtime.h>

typedef float v2f __attribute__((ext_vector_type(2)));
typedef float v4f __attribute__((ext_vector_type(4)));
typedef float v8f __attribute__((ext_vector_type(8)));

#define LEECH_EPS 1e-8f

// ds_swizzle group-of-32: src = ((lane & 0x1f) | 0) ^ mask  -> 1-instr xor butterfly
#define SWZ_XOR(v, m) \
  __int_as_float(__builtin_amdgcn_ds_swizzle(__float_as_int(v), 0x1f | ((m) << 10)))

// workspace layout (float offsets)
#define WENC_OFF  0        // padded W_enc  [32][1024]
#define WDEC_OFF  32768    // padded W_dec  [1024][32]
#define BENC_OFF  65536    // padded b_enc  [32]
#define G_OFF     65568    // Gram = W_dec^T W_dec, padded [32][32]
#define U_OFF     66592    // u = W_dec^T b_dec, padded [32]
#define BETA_OFF  66624    // ||b_dec||^2
#define GPART_OFF 66640    // partial sums [16][1057]
#define NPP    1057
#define NSLICE 16

// ---------------- prep: pad weights into WMMA-friendly buffers ----------------
__global__ __launch_bounds__(256) void leech_prep_pad(
    const float* __restrict__ wenc, const float* __restrict__ wdec,
    const float* __restrict__ benc, float* __restrict__ ws)
{
  int i = blockIdx.x * 256 + threadIdx.x;
  if (i < 32768) {
    int row = i >> 10, c = i & 1023;
    ws[WENC_OFF + i] = (row < 24) ? wenc[row * 1024 + c] : 0.0f;
  } else if (i < 65536) {
    int j = i - 32768;
    int d = j >> 5, l = j & 31;
    ws[WDEC_OFF + j] = (l < 24) ? wdec[d * 24 + l] : 0.0f;
  } else if (i < 65568) {
    int l = i - 65536;
    ws[BENC_OFF + l] = (l < 24) ? benc[l] : 0.0f;
  }
}

// ---------------- prep: Gram matrix partials (deterministic 2-stage) ----------
__global__ __launch_bounds__(256) void leech_gram_part(
    const float* __restrict__ wdec, const float* __restrict__ bdec,
    float* __restrict__ ws)
{
  int gid = blockIdx.x * 256 + threadIdx.x;
  if (gid >= NPP * NSLICE) return;
  int pp = gid >> 4;
  int slice = gid & 15;
  int d0 = slice * 64;
  float acc = 0.0f;
  if (pp < 1024) {
    int l = pp >> 5, m = pp & 31;
    if (l < 24 && m < 24)
      for (int d = d0; d < d0 + 64; ++d)
        acc += wdec[d * 24 + l] * wdec[d * 24 + m];
  } else if (pp < 1056) {
    int l = pp - 1024;
    if (l < 24)
      for (int d = d0; d < d0 + 64; ++d)
        acc += wdec[d * 24 + l] * bdec[d];
  } else {
    for (int d = d0; d < d0 + 64; ++d)
      acc += bdec[d] * bdec[d];
  }
  ws[GPART_OFF + slice * NPP + pp] = acc;
}

__global__ __launch_bounds__(256) void leech_gram_reduce(float* __restrict__ ws)
{
  int pp = blockIdx.x * 256 + threadIdx.x;
  if (pp >= NPP) return;
  float s = 0.0f;
  for (int sl = 0; sl < NSLICE; ++sl) s += ws[GPART_OFF + sl * NPP + pp];
  if (pp < 1024)      ws[G_OFF + pp] = s;
  else if (pp < 1056) ws[U_OFF + pp - 1024] = s;
  else                ws[BETA_OFF] = s;
}

// ---------------- fused main kernel: one 16-token tile per wave ---------------
__global__ __launch_bounds__(256) void leech_main(
    const float* __restrict__ data,   // [32768][1024]
    const float* __restrict__ wencp,  // padded [32][1024]
    const float* __restrict__ bencp,  // padded [32]
    const float* __restrict__ wdecp,  // padded [1024][32]
    const float* __restrict__ bdec,   // [1024]
    const float* __restrict__ G,      // padded [32][32]
    const float* __restrict__ U,      // padded [32]
    const float* __restrict__ betap,
    const float* __restrict__ ecsp, const float* __restrict__ epp,
    float* __restrict__ out)          // [32768][1024]
{
  __shared__ float corr[8][16][32];   // per-wave corrected-lattice tile
  __shared__ float sc[8][16];         // per-wave ||lattice|| per token
  const int lane = threadIdx.x & 31;
  const int w    = threadIdx.x >> 5;
  const int col  = lane & 15;
  const int hi   = lane >> 4;
  const int tile = blockIdx.x * 8 + w;
  const int tok0 = tile * 16;
  const float ecs = ecsp[0], ep = epp[0];

  // ---- encode GEMM: proj[16 tok x 32 lat] over K=1024, + ||data|| on the fly
  v8f c0 = {}; v8f c1 = {};
  float in_sq = 0.0f;
  const float* arow = data  + (size_t)(tok0 + col) * 1024 + 2 * hi;
  const float* b0p  = wencp + (size_t)col        * 1024 + 2 * hi;
  const float* b1p  = wencp + (size_t)(col + 16) * 1024 + 2 * hi;
  #pragma unroll 4
  for (int k = 0; k < 1024; k += 4) {
    v2f a  = __builtin_nontemporal_load((const v2f*)(arow + k));  // streamed once
    v2f b0 = *(const v2f*)(b0p + k);
    v2f b1 = *(const v2f*)(b1p + k);
    in_sq += a.x * a.x + a.y * a.y;
    c0 = __builtin_amdgcn_wmma_f32_16x16x4_f32(false, a, false, b0, (short)0, c0, false, false);
    c1 = __builtin_amdgcn_wmma_f32_16x16x4_f32(false, a, false, b1, (short)0, c1, false, false);
  }
  in_sq += SWZ_XOR(in_sq, 16);        // full ||data[tok=col]||^2 in all lanes

  // ---- bias + lattice quantize; ||lattice_points|| via 1-instr butterflies
  const float be0 = bencp[col], be1 = bencp[col + 16];
  float q0[8], q1[8], s1[8], ine2[8];
  #pragma unroll
  for (int r = 0; r < 8; ++r) {
    q0[r] = rintf((c0[r] + be0) / ecs) * ecs;   // RNE like jnp.round
    q1[r] = rintf((c1[r] + be1) / ecs) * ecs;
  }
  const int bperm_base = hi * 32;               // byte index base for bpermute
  #pragma unroll
  for (int r = 0; r < 8; ++r) {
    float p = q0[r] * q0[r] + q1[r] * q1[r];    // token = r + 8*hi
    p += SWZ_XOR(p, 1);
    p += SWZ_XOR(p, 2);
    p += SWZ_XOR(p, 4);
    p += SWZ_XOR(p, 8);
    float oute = sqrtf(p);
    float insq_t = __int_as_float(
        __builtin_amdgcn_ds_bpermute(bperm_base + 4 * r, __float_as_int(in_sq)));
    s1[r]   = sqrtf(insq_t) / (oute + LEECH_EPS) * ep;
    ine2[r] = s1[r] * oute;                     // ||lattice|| (s1 >= 0)
  }

  // ---- corrected lattice -> LDS; ||lattice|| -> token-indexed LDS
  #pragma unroll
  for (int r = 0; r < 8; ++r) {
    float l0 = q0[r] * s1[r], l1 = q1[r] * s1[r];
    int t = r + 8 * hi;
    corr[w][t][col]      = (fabsf(l0) > ecs) ? l0 : 0.0f;
    corr[w][t][col + 16] = (fabsf(l1) > ecs) ? l1 : 0.0f;
    if (col == r) sc[w][t] = ine2[r];
  }
  __syncthreads();

  // ---- chunk-invariant corrected fragments (B-matrix: c^T, K=lat, N=token)
  v2f bfr[8];
  #pragma unroll
  for (int j = 0; j < 8; ++j)
    bfr[j] = *(const v2f*)&corr[w][col][4 * j + 2 * hi];

  // ---- ||res||^2 via Gram, flipped: C' = G_rows x c^T  (token-per-lane out)
  v8f t0 = {}; v8f t1 = {};
  #pragma unroll
  for (int j = 0; j < 8; ++j) {
    v2f g0 = *(const v2f*)(G + (size_t)col        * 32 + 4 * j + 2 * hi);
    v2f g1 = *(const v2f*)(G + (size_t)(col + 16) * 32 + 4 * j + 2 * hi);
    t0 = __builtin_amdgcn_wmma_f32_16x16x4_f32(false, g0, false, bfr[j], (short)0, t0, false, false);
    t1 = __builtin_amdgcn_wmma_f32_16x16x4_f32(false, g1, false, bfr[j], (short)0, t1, false, false);
  }
  // c[t][l'] for this lane's latent halves, contiguous b128 LDS loads
  v4f ca0 = *(const v4f*)&corr[w][col][8 * hi];
  v4f ca1 = *(const v4f*)&corr[w][col][8 * hi + 4];
  v4f cb0 = *(const v4f*)&corr[w][col][16 + 8 * hi];
  v4f cb1 = *(const v4f*)&corr[w][col][16 + 8 * hi + 4];
  v4f ua0 = *(const v4f*)(U + 8 * hi);
  v4f ua1 = *(const v4f*)(U + 8 * hi + 4);
  v4f ub0 = *(const v4f*)(U + 16 + 8 * hi);
  v4f ub1 = *(const v4f*)(U + 16 + 8 * hi + 4);
  float p = 0.0f;
  #pragma unroll
  for (int r = 0; r < 4; ++r) {
    p += t0[r]     * ca0[r] + t0[r + 4] * ca1[r];
    p += t1[r]     * cb0[r] + t1[r + 4] * cb1[r];
    p += 2.0f * (ua0[r] * ca0[r] + ua1[r] * ca1[r]
               + ub0[r] * cb0[r] + ub1[r] * cb1[r]);
  }
  p += SWZ_XOR(p, 16);                          // sum both latent halves
  float g = p + betap[0];                       // ||res||^2 for token = col
  float s2 = (sc[w][col] / (sqrtf(g) + LEECH_EPS)) * ep;  // per-lane scalar

  // ---- decode GEMM flipped: C = W_dec_rows x c^T -> lane holds 8 consecutive d
  for (int ch = 0; ch < 64; ++ch) {
    const float* wr = wdecp + (size_t)(ch * 16 + col) * 32 + 2 * hi;
    v8f acc = {};
    #pragma unroll
    for (int j = 0; j < 8; ++j) {
      v2f a = *(const v2f*)(wr + 4 * j);
      acc = __builtin_amdgcn_wmma_f32_16x16x4_f32(false, a, false, bfr[j], (short)0, acc, false, false);
    }
    const int dbase = ch * 16 + 8 * hi;         // d = dbase + r  (r = 0..7)
    v4f bd0 = *(const v4f*)(bdec + dbase);
    v4f bd1 = *(const v4f*)(bdec + dbase + 4);
    v4f o0, o1;
    #pragma unroll
    for (int r = 0; r < 4; ++r) {
      o0[r] = (acc[r]     + bd0[r]) * s2;
      o1[r] = (acc[r + 4] + bd1[r]) * s2;
    }
    float* op = out + (size_t)(tok0 + col) * 1024 + dbase;
    __builtin_nontemporal_store(o0, (v4f*)op);
    __builtin_nontemporal_store(o1, (v4f*)(op + 4));
  }
}

// ------------------------------- launcher -------------------------------------
extern "C" void kernel_launch(void* const* d_in, const int* in_sizes, int n_in,
                              void* d_out, int out_size, void* d_ws, size_t ws_size,
                              hipStream_t stream)
{
  const float* data = (const float*)d_in[0];
  const float* wenc = (const float*)d_in[1];
  const float* benc = (const float*)d_in[2];
  const float* wdec = (const float*)d_in[3];
  const float* bdec = (const float*)d_in[4];
  const float* ecs  = (const float*)d_in[5];
  const float* ep   = (const float*)d_in[6];
  float* out = (float*)d_out;
  float* ws  = (float*)d_ws;

  leech_prep_pad<<<257, 256, 0, stream>>>(wenc, wdec, benc, ws);
  leech_gram_part<<<(NPP * NSLICE + 255) / 256, 256, 0, stream>>>(wdec, bdec, ws);
  leech_gram_reduce<<<(NPP + 255) / 256, 256, 0, stream>>>(ws);
  leech_main<<<256, 256, 0, stream>>>(
      data, ws + WENC_OFF, ws + BENC_OFF, ws + WDEC_OFF, bdec,
      ws + G_OFF, ws + U_OFF, ws + BETA_OFF, ecs, ep, out);
}